// Cace_84490596647230
// MI455X (gfx1250) — compile-verified
//
#include <hip/hip_runtime.h>
#include <hip/hip_bf16.h>
#include <math.h>

// ---------------------------------------------------------------------------
// CACE-like edge featurization + 1 message-passing step for MI455X (gfx1250).
// Scatter/gather dominated; A-feature tensors (2 x 34.6MB) are L2-resident
// (192MB L2), so all atomics/gathers stay on-chip. The radial transform
// (batched (16x6)x(6x6) GEMM) runs on V_WMMA_F32_16X16X4_F32.
// ---------------------------------------------------------------------------

typedef float v2f __attribute__((ext_vector_type(2)));
typedef float v8f __attribute__((ext_vector_type(8)));

#define N_RBF   6
#define N_ANG   20
#define N_P     6
#define N_C     9
#define NPAC    1080          // N_P * N_ANG * N_C
#define CUTOFFF 5.5f
#define EPSF    1e-9f
#define MPNORM  0.31622776601683794f   // 1/sqrt(10)

__device__ __constant__ int   d_LX[N_ANG]   = {0, 1,0,0, 2,1,1,0,0,0, 3,2,2,1,1,1,0,0,0,0};
__device__ __constant__ int   d_LY[N_ANG]   = {0, 0,1,0, 0,1,0,2,1,0, 0,1,0,2,1,0,3,2,1,0};
__device__ __constant__ int   d_LZ[N_ANG]   = {0, 0,0,1, 0,0,1,0,1,2, 0,0,1,0,1,2,0,1,2,3};
__device__ __constant__ int   d_LSUM[N_ANG] = {0, 1,1,1, 2,2,2,2,2,2, 3,3,3,3,3,3,3,3,3,3};
__device__ __constant__ float d_PREF[N_ANG] = {1.f, 1.f,1.f,1.f,
                                               1.f,2.f,2.f,1.f,2.f,1.f,
                                               1.f,3.f,3.f,3.f,6.f,3.f,1.f,3.f,3.f,1.f};
__device__ __constant__ int   d_ZS[3] = {1, 6, 8};

// ---------------------------------------------------------------------------
// 0) zero a float buffer (vectorized; n multiple of 4 guaranteed by caller)
__global__ void zero_kernel(float4* __restrict__ buf, int n4) {
    int i = blockIdx.x * blockDim.x + threadIdx.x;
    if (i < n4) buf[i] = make_float4(0.f, 0.f, 0.f, 0.f);
}

// ---------------------------------------------------------------------------
// 1) per-edge geometry: S = rad*fcut (E x 6), fcut (E), ang (E x 20), enc (E x 9)
__global__ void edge_geom_kernel(const float* __restrict__ pos,
                                 const int*   __restrict__ zatom,
                                 const int*   __restrict__ eidx,
                                 const float* __restrict__ shifts,
                                 const float* __restrict__ embW,   // (3 species, 3 basis)
                                 float* __restrict__ S,
                                 float* __restrict__ Fc,
                                 float* __restrict__ ANG,
                                 float* __restrict__ ENC,
                                 int E) {
    int e = blockIdx.x * blockDim.x + threadIdx.x;
    if (e >= E) return;
    int s = eidx[e];
    int t = eidx[E + e];

    float vx = pos[3*t+0] - pos[3*s+0] + shifts[3*e+0];
    float vy = pos[3*t+1] - pos[3*s+1] + shifts[3*e+1];
    float vz = pos[3*t+2] - pos[3*s+2] + shifts[3*e+2];
    float len  = sqrtf(vx*vx + vy*vy + vz*vz);
    float inv  = 1.0f / (len + EPSF);
    float ux = vx * inv, uy = vy * inv, uz = vz * inv;

    // cutoff polynomial (p=6): 1 - 28 u^6 + 48 u^7 - 21 u^8, u<1
    float u  = len / CUTOFFF;
    float fc = 0.0f;
    if (u < 1.0f) {
        float u2 = u*u, u3 = u2*u, u6 = u3*u3;
        fc = 1.0f - 28.0f*u6 + 48.0f*u6*u - 21.0f*u6*u2;
    }
    Fc[e] = fc;

    // radial basis * cutoff
    const float norm = 0.60302268915552726f;   // sqrt(2/5.5)
    const float kpi  = 3.14159265358979323846f / CUTOFFF;
    #pragma unroll
    for (int r = 0; r < N_RBF; ++r) {
        float rad = norm * sinf((float)(r+1) * kpi * len) * inv;
        S[e*N_RBF + r] = rad * fc;
    }

    // angular monomials: ux^lx * uy^ly * uz^lz (0^0 == 1)
    float px[4] = {1.f, ux, ux*ux, ux*ux*ux};
    float py[4] = {1.f, uy, uy*uy, uy*uy*uy};
    float pz[4] = {1.f, uz, uz*uz, uz*uz*uz};
    #pragma unroll
    for (int a = 0; a < N_ANG; ++a)
        ANG[e*N_ANG + a] = px[d_LX[a]] * py[d_LY[a]] * pz[d_LZ[a]];

    // species embedding outer product
    int zs = zatom[s], zt = zatom[t];
    int ss = 0, st = 0;
    #pragma unroll
    for (int k = 0; k < 3; ++k) { if (zs == d_ZS[k]) ss = k; if (zt == d_ZS[k]) st = k; }
    float emS[3] = {embW[3*ss+0], embW[3*ss+1], embW[3*ss+2]};
    float emT[3] = {embW[3*st+0], embW[3*st+1], embW[3*st+2]};
    #pragma unroll
    for (int i = 0; i < 3; ++i)
        #pragma unroll
        for (int j = 0; j < 3; ++j)
            ENC[e*N_C + i*3 + j] = emS[i] * emT[j];
}

// ---------------------------------------------------------------------------
// 2) radial transform via FP32 WMMA: R[e,l,p] = sum_r S[e,r] * W[l][r,p]
//    One wave handles a 16-edge tile; A = S tile (16x6, K padded to 8 via two
//    16x16x4 WMMAs), B = W[l] (6x6 zero-padded into 4x16 K-chunks).
//    Divergence-free operand setup (selects, not exec-masked loads); the
//    full-tile / partial-tile split is wave-uniform.
__global__ void radial_wmma_kernel(const float* __restrict__ S,
                                   const float* __restrict__ W,   // (4,6,6)
                                   float* __restrict__ R,          // (E,4,6)
                                   int E) {
    int lane = threadIdx.x & 31;
    int tile = blockIdx.x * (blockDim.x >> 5) + (threadIdx.x >> 5);
    int ntiles = (E + 15) >> 4;
    if (tile >= ntiles) return;                 // wave-uniform

    const bool fullTile = (tile * 16 + 16) <= E;   // wave-uniform
    int half = lane >> 4;                  // 0: K lanes 0..1 / D rows 0..7; 1: K 2..3 / rows 8..15
    int m    = lane & 15;
    int e    = tile * 16 + m;
    int p    = m;                          // N index of B / D
    int pc   = (p < N_P) ? p : 0;          // clamped, always in-bounds
    bool pOK = (p < N_P);
    bool lo  = (half == 0);

    // ---- A operand: lane m = edge row, vgpr j -> K = 2*half + j
    v2f a0 = {0.f, 0.f};                   // K 0..3 chunk
    v2f a1 = {0.f, 0.f};                   // K 4..7 chunk (only 4,5 populated)
    if (fullTile) {
        const float* Se = S + e * N_RBF;
        a0.x = Se[2*half + 0];
        a0.y = Se[2*half + 1];
        float s4 = Se[4], s5 = Se[5];
        a1.x = lo ? s4 : 0.f;
        a1.y = lo ? s5 : 0.f;
    } else if (e < E) {
        const float* Se = S + e * N_RBF;
        a0.x = Se[2*half + 0];
        a0.y = Se[2*half + 1];
        if (lo) { a1.x = Se[4]; a1.y = Se[5]; }
    }

    for (int l = 0; l < 4; ++l) {
        const float* Wl = W + l * 36;
        // ---- B operand: vgpr j -> K = 2*half + j, N = lane%16 (selects, no exec masking)
        float w0 = Wl[(2*half + 0)*N_P + pc];
        float w1 = Wl[(2*half + 1)*N_P + pc];
        float w4 = Wl[4*N_P + pc];
        float w5 = Wl[5*N_P + pc];
        v2f b0, b1;
        b0.x = pOK ? w0 : 0.f;
        b0.y = pOK ? w1 : 0.f;
        b1.x = (pOK && lo) ? w4 : 0.f;
        b1.y = (pOK && lo) ? w5 : 0.f;

        v8f c = {};
        c = __builtin_amdgcn_wmma_f32_16x16x4_f32(false, a0, false, b0,
                                                  (short)0, c, false, false);
        c = __builtin_amdgcn_wmma_f32_16x16x4_f32(false, a1, false, b1,
                                                  (short)0, c, false, false);

        // ---- D: vgpr v -> row M = v + 8*half, col N = lane%16
        if (fullTile) {
            if (pOK) {
                int ebase = tile * 16 + 8*half;
                #pragma unroll
                for (int v = 0; v < 8; ++v)
                    R[((ebase + v)*4 + l)*N_P + p] = c[v];
            }
        } else {
            #pragma unroll
            for (int v = 0; v < 8; ++v) {
                int ee = tile * 16 + v + 8*half;
                if (pOK && ee < E) R[(ee*4 + l)*N_P + p] = c[v];
            }
        }
    }
}

// ---------------------------------------------------------------------------
// 3) edge scatter: A0[t, p, a, c] += R[e, lsum(a), p] * ang[e,a] * enc[e,c]
//    thread per (edge, p) -> 180 f32 atomics (L2-resident target)
__global__ void scatter_kernel(const int*   __restrict__ eidx,
                               const float* __restrict__ Fc,
                               const float* __restrict__ ANG,
                               const float* __restrict__ ENC,
                               const float* __restrict__ R,
                               float* __restrict__ A0,
                               int E) {
    int idx = blockIdx.x * blockDim.x + threadIdx.x;
    if (idx >= E * N_P) return;
    int e = idx / N_P;
    int p = idx - e * N_P;
    if (Fc[e] == 0.0f) return;            // S==0 -> R==0 -> nothing to add

    int t = eidx[E + e];
    float Rl[4];
    #pragma unroll
    for (int l = 0; l < 4; ++l) Rl[l] = R[(e*4 + l)*N_P + p];
    float enc[N_C];
    #pragma unroll
    for (int c = 0; c < N_C; ++c) enc[c] = ENC[e*N_C + c];

    float* base = A0 + (size_t)t * NPAC + p * (N_ANG * N_C);
    #pragma unroll
    for (int a = 0; a < N_ANG; ++a) {
        float w = Rl[d_LSUM[a]] * ANG[e*N_ANG + a];
        #pragma unroll
        for (int c = 0; c < N_C; ++c)
            atomicAdd(base + a*N_C + c, w * enc[c]);
    }
}

// ---------------------------------------------------------------------------
// 4) A1 = memory_coef * A0 (vectorized)
__global__ void blend_kernel(const float4* __restrict__ A0,
                             float4* __restrict__ A1,
                             const float* __restrict__ mcoef,
                             int n4) {
    int i = blockIdx.x * blockDim.x + threadIdx.x;
    if (i >= n4) return;
    float m = mcoef[0];
    float4 v = A0[i];
    A1[i] = make_float4(m*v.x, m*v.y, m*v.z, m*v.w);
}

// ---------------------------------------------------------------------------
// 5) message passing: A1[t,:] += mp_norm * fcut[e] * A0[s,:]   (block per edge)
//    b128 loads (1080 = 270 x float4); f32 atomic adds into L2.
__global__ void mp_kernel(const int*   __restrict__ eidx,
                          const float* __restrict__ Fc,
                          const float* __restrict__ A0,
                          float* __restrict__ A1,
                          int E) {
    int e = blockIdx.x;
    float f = Fc[e] * MPNORM;
    if (f == 0.0f) return;                // zero contribution
    int s = eidx[e];
    int t = eidx[E + e];
    const float4* src = (const float4*)(A0 + (size_t)s * NPAC);
    float*        dst = A1 + (size_t)t * NPAC;
    for (int k = threadIdx.x; k < NPAC/4; k += blockDim.x) {
        float4 v = src[k];
        atomicAdd(dst + 4*k + 0, f * v.x);
        atomicAdd(dst + 4*k + 1, f * v.y);
        atomicAdd(dst + 4*k + 2, f * v.z);
        atomicAdd(dst + 4*k + 3, f * v.w);
    }
}

// ---------------------------------------------------------------------------
// 6) symmetrize both features, interleave into output (N, 6, 5, 9, 2)
__global__ void out_kernel(const float* __restrict__ A0,
                           const float* __restrict__ A1,
                           float* __restrict__ out,
                           int N) {
    int idx = blockIdx.x * blockDim.x + threadIdx.x;
    if (idx >= N * N_P * N_C) return;
    int c   = idx % N_C;
    int tmp = idx / N_C;
    int p   = tmp % N_P;
    int n   = tmp / N_P;

    size_t ob = ((((size_t)n * N_P + p) * 5) * N_C + c) * 2;   // j=0, f=0
    #pragma unroll
    for (int f = 0; f < 2; ++f) {
        const float* Af = (f ? A1 : A0) + (size_t)n * NPAC + p * (N_ANG * N_C);
        float B2[4] = {0.f, 0.f, 0.f, 0.f};
        #pragma unroll
        for (int a = 0; a < N_ANG; ++a) {
            float v = Af[a*N_C + c];
            B2[d_LSUM[a]] += d_PREF[a] * v * v;
        }
        out[ob + f] = Af[c];              // B1 (a = 0 slice)
        #pragma unroll
        for (int l = 0; l < 4; ++l)
            out[ob + (size_t)(l + 1) * (N_C * 2) + f] = B2[l];
    }
}

// ---------------------------------------------------------------------------
extern "C" void kernel_launch(void* const* d_in, const int* in_sizes, int n_in,
                              void* d_out, int out_size, void* d_ws, size_t ws_size,
                              hipStream_t stream) {
    const float* pos    = (const float*)d_in[0];
    const int*   zatom  = (const int*)  d_in[1];
    const int*   eidx   = (const int*)  d_in[2];
    const float* shifts = (const float*)d_in[3];
    const float* embW   = (const float*)d_in[4];
    const float* radW   = (const float*)d_in[5];   // (4, 6, 6)
    const float* mcoef  = (const float*)d_in[6];
    float*       out    = (float*)d_out;

    const int N = in_sizes[0] / 3;
    const int E = in_sizes[2] / 2;

    // workspace layout (floats); base is 16B-aligned, NPAC*4B is 16B-multiple
    float* ws  = (float*)d_ws;
    float* A0  = ws;                         size_t off = (size_t)N * NPAC;
    float* A1  = ws + off;                   off += (size_t)N * NPAC;
    float* S   = ws + off;                   off += (size_t)E * N_RBF;
    float* Fc  = ws + off;                   off += (size_t)E;
    float* ANG = ws + off;                   off += (size_t)E * N_ANG;
    float* ENC = ws + off;                   off += (size_t)E * N_C;
    float* R   = ws + off;                   off += (size_t)E * 4 * N_P;

    const int BS = 256;
    int nA  = N * NPAC;        // multiple of 4 (NPAC = 1080)
    int nA4 = nA / 4;

    // 0) zero accumulator
    zero_kernel<<<(nA4 + BS - 1) / BS, BS, 0, stream>>>((float4*)A0, nA4);

    // 1) per-edge geometry
    edge_geom_kernel<<<(E + BS - 1) / BS, BS, 0, stream>>>(
        pos, zatom, eidx, shifts, embW, S, Fc, ANG, ENC, E);

    // 2) WMMA radial transform (16 edges per wave, 8 waves per block)
    int ntiles = (E + 15) / 16;
    radial_wmma_kernel<<<(ntiles + 7) / 8, BS, 0, stream>>>(S, radW, R, E);

    // 3) edge scatter into A0
    scatter_kernel<<<(E * N_P + BS - 1) / BS, BS, 0, stream>>>(
        eidx, Fc, ANG, ENC, R, A0, E);

    // 4) A1 = memory_coef * A0
    blend_kernel<<<(nA4 + BS - 1) / BS, BS, 0, stream>>>(
        (const float4*)A0, (float4*)A1, mcoef, nA4);

    // 5) message passing
    mp_kernel<<<E, BS, 0, stream>>>(eidx, Fc, A0, A1, E);

    // 6) symmetrize + write output
    int nOut = N * N_P * N_C;
    out_kernel<<<(nOut + BS - 1) / BS, BS, 0, stream>>>(A0, A1, out, N);
}